// GraphAttentionEmbedding_25958782337590
// MI455X (gfx1250) — compile-verified
//
#include <hip/hip_runtime.h>

typedef __attribute__((ext_vector_type(2))) float v2f;
typedef __attribute__((ext_vector_type(8))) float v8f;

#define B_ROWS 4096
#define SEQ_L  128
#define DIM    64

// ---------------------------------------------------------------------------
// Stage 1: per-row co-occurrence counts + per-position MLP layer-1 + weighted
// pooling, producing G[(src rows 0..4095, dst rows 4096..8191), 64].
// One block = one batch row. 128 threads (4 wave32 waves).
// ---------------------------------------------------------------------------
__global__ __launch_bounds__(128) void pool_counts_kernel(
    const int*   __restrict__ src_ids,  // (B, L)
    const int*   __restrict__ dst_ids,  // (B, L)
    const float* __restrict__ W1,       // (1, D) -> D floats
    const float* __restrict__ b1,       // (D,)
    float*       __restrict__ G)        // (2*B, D) workspace
{
    __shared__ int s_src[SEQ_L];
    __shared__ int s_dst[SEQ_L];
    __shared__ int s_css[SEQ_L], s_csd[SEQ_L], s_cds[SEQ_L], s_cdd[SEQ_L];

    const int b = blockIdx.x;
    const int t = threadIdx.x;

    const int sid = src_ids[b * SEQ_L + t];
    const int did = dst_ids[b * SEQ_L + t];
    s_src[t] = sid;
    s_dst[t] = did;
    __syncthreads();

    // 4 pairwise-equality counts for this position (hash-join as compares)
    int css = 0, csd = 0, cds = 0, cdd = 0;
#pragma unroll 8
    for (int j = 0; j < SEQ_L; ++j) {
        const int sj = s_src[j];
        const int dj = s_dst[j];
        css += (sj == sid);
        csd += (dj == sid);
        cds += (sj == did);
        cdd += (dj == did);
    }
    // padding mask: id==0 zeroes both channels for that position
    if (sid == 0) { css = 0; csd = 0; }
    if (did == 0) { cds = 0; cdd = 0; }
    s_css[t] = css; s_csd[t] = csd; s_cds[t] = cds; s_cdd[t] = cdd;
    __syncthreads();

    // Phase 2: threads 0..63 build the src-pooled row, 64..127 the dst row.
    // G[row,k] = sum_l (w_l / sum_w) * (relu(c0*W1[k]+b1[k]) + relu(c1*W1[k]+b1[k]))
    // sum_w = sum_{l=0}^{127} e^{l-128} = (1 - e^-128) / (e - 1)
    const float inv_sumw = (expf(1.0f) - 1.0f) / (1.0f - expf(-128.0f));
    const int  k     = t & 63;
    const bool isDst = (t >= 64);
    const float w1k = W1[k];
    const float b1k = b1[k];
    const int* c0 = isDst ? s_cds : s_css;
    const int* c1 = isDst ? s_cdd : s_csd;

    float acc = 0.0f;
#pragma unroll 4
    for (int i = 0; i < SEQ_L; ++i) {
        const float w  = __expf((float)(i - SEQ_L)) * inv_sumw;
        const float h0 = fmaxf((float)c0[i] * w1k + b1k, 0.0f);
        const float h1 = fmaxf((float)c1[i] * w1k + b1k, 0.0f);
        acc += w * (h0 + h1);
    }
    const int row = b + (isDst ? B_ROWS : 0);
    G[row * DIM + k] = acc;
}

// ---------------------------------------------------------------------------
// Stage 2: out = G @ W2 + 2*b2  via V_WMMA_F32_16X16X4_F32.
// (8192 x 64) @ (64 x 64). One wave per 16x16 output tile; K-loop = 16 x K4.
// 2048 tiles -> 256 blocks of 256 threads (8 wave32 waves).
// ---------------------------------------------------------------------------
__global__ __launch_bounds__(256) void wmma_gemm_kernel(
    const float* __restrict__ G,    // (2*B, 64) row-major
    const float* __restrict__ W2,   // (64, 64) row-major [k][n]
    const float* __restrict__ b2,   // (64,)
    float*       __restrict__ out)  // (2*B, 64) = d_out
{
    const int lane = threadIdx.x & 31;   // wave32
    const int wave = threadIdx.x >> 5;
    const int tile = blockIdx.x * 8 + wave;
    const int tile_m = tile >> 2;        // 512 row-tiles
    const int tile_n = tile & 3;         // 4 col-tiles

    const int mBase = tile_m * 16;
    const int nBase = tile_n * 16;

    // f32 WMMA A(16x4): lane holds M = lane%16, K = 2*(lane/16) + {0,1}
    // B(4x16) mirrors with N across lanes.
    const int mn    = lane & 15;
    const int khalf = (lane >> 4) << 1;

    v8f c = {};
#pragma unroll
    for (int s = 0; s < 16; ++s) {
        const int kb = s * 4 + khalf;
        v2f a, bm;
        a.x  = G[(mBase + mn) * DIM + kb];
        a.y  = G[(mBase + mn) * DIM + kb + 1];
        bm.x = W2[kb * DIM + (nBase + mn)];
        bm.y = W2[(kb + 1) * DIM + (nBase + mn)];
        c = __builtin_amdgcn_wmma_f32_16x16x4_f32(
                /*neg_a=*/false, a, /*neg_b=*/false, bm,
                /*c_mod=*/(short)0, c, /*reuse_a=*/false, /*reuse_b=*/false);
    }

    // C/D 16x16 f32: VGPR r holds M = r + 8*(lane/16), N = lane%16
    const float bias = 2.0f * b2[nBase + mn];
    const int mOff = (lane >> 4) << 3;
#pragma unroll
    for (int r = 0; r < 8; ++r) {
        out[(mBase + mOff + r) * DIM + (nBase + mn)] = c[r] + bias;
    }
}

extern "C" void kernel_launch(void* const* d_in, const int* in_sizes, int n_in,
                              void* d_out, int out_size, void* d_ws, size_t ws_size,
                              hipStream_t stream) {
    const int*   src_ids = (const int*)  d_in[0];
    const int*   dst_ids = (const int*)  d_in[1];
    const float* W1      = (const float*)d_in[2];
    const float* b1      = (const float*)d_in[3];
    const float* W2      = (const float*)d_in[4];
    const float* b2      = (const float*)d_in[5];

    float* G   = (float*)d_ws;   // (8192, 64) f32 = 2 MB
    float* out = (float*)d_out;  // src_feat rows 0..4095, dst_feat rows 4096..8191

    pool_counts_kernel<<<B_ROWS, 128, 0, stream>>>(src_ids, dst_ids, W1, b1, G);

    const int tiles  = (2 * B_ROWS / 16) * (DIM / 16); // 2048
    const int blocks = tiles / 8;                      // 8 waves per block
    wmma_gemm_kernel<<<blocks, 256, 0, stream>>>(G, W2, b2, out);
}